// NodeGat_82910048682338
// MI455X (gfx1250) — compile-verified
//
#include <hip/hip_runtime.h>
#include <math.h>

#define F_DIM 128
#define H_HEADS 4
#define C_HID 32
#define NCLS 40
#define NEG_SLOPE 0.2f
#define EPS_V 1e-16f

typedef float v2f __attribute__((ext_vector_type(2)));
typedef float v8f __attribute__((ext_vector_type(8)));

// ---------------------------------------------------------------- fill
__global__ void fill_kernel(float* p, float v, int n) {
  int i = blockIdx.x * blockDim.x + threadIdx.x;
  if (i < n) p[i] = v;
}

// ---------------------------------------------------------------- WMMA GEMM
// Hout[nrows x 128] = X[nrows x 128] @ W[128 x 128], fp32 via V_WMMA_F32_16X16X4_F32.
// grid.x = ceil(nrows/16) tiles; block = 256 threads = 8 waves; wave w -> cols [16w,16w+16).
__global__ void gemm128_wmma(const float* __restrict__ X,
                             const float* __restrict__ W,
                             float* __restrict__ Hout, int nrows) {
  __shared__ float As[16 * 128];            // 16-row tile of X, row-major
  const int row0 = blockIdx.x * 16;
  const int t = threadIdx.x;
  const bool full_tile = (row0 + 16 <= nrows);   // uniform across block
  if (full_tile) {
#pragma unroll
    for (int i = 0; i < 8; ++i) {           // 256 thr * 8 = 2048 floats
      int idx = t + i * 256;
      As[idx] = X[(row0 + (idx >> 7)) * 128 + (idx & 127)];
    }
  } else {
#pragma unroll
    for (int i = 0; i < 8; ++i) {
      int idx = t + i * 256;
      int gr = row0 + (idx >> 7);
      As[idx] = (gr < nrows) ? X[gr * 128 + (idx & 127)] : 0.0f;
    }
  }
  __syncthreads();

  const int lane = t & 31;
  const int col0 = (t >> 5) * 16;           // wave id * 16
  const int m    = lane & 15;               // A row / B column / D column index
  const int kh   = (lane & 16) ? 2 : 0;     // K offset of this lane half

  v8f acc = {0.f, 0.f, 0.f, 0.f, 0.f, 0.f, 0.f, 0.f};
#pragma unroll 4
  for (int kb = 0; kb < 128; kb += 4) {
    v2f a, b;
    // A 16x4: VGPR0 = A[m][kb+kh], VGPR1 = A[m][kb+kh+1]   (kh = 0|2 per lane half)
    a.x = As[m * 128 + kb + kh];
    a.y = As[m * 128 + kb + kh + 1];
    // B 4x16: VGPR0 = B[kb+kh][col], VGPR1 = B[kb+kh+1][col]
    b.x = W[(kb + kh) * 128 + col0 + m];
    b.y = W[(kb + kh + 1) * 128 + col0 + m];
    acc = __builtin_amdgcn_wmma_f32_16x16x4_f32(false, a, false, b,
                                                (short)0, acc, false, false);
  }
  // D 16x16 f32: VGPR r -> M = r (lanes 0-15) | r+8 (lanes 16-31), N = lane%16
  const int mo = (lane & 16) ? 8 : 0;
  if (full_tile) {
#pragma unroll
    for (int r = 0; r < 8; ++r)
      Hout[(row0 + r + mo) * 128 + col0 + m] = acc[r];
  } else {
#pragma unroll
    for (int r = 0; r < 8; ++r) {
      int gr = row0 + r + mo;
      if (gr < nrows) Hout[gr * 128 + col0 + m] = acc[r];
    }
  }
}

// ---------------------------------------------------------------- per-node attn coeffs
// al_s[n,h] = sum_c h[n,h,c]*a_src[h,c]; one thread per (n,head)
__global__ void node_al_kernel(const float* __restrict__ Hfeat,
                               const float* __restrict__ a_src,
                               const float* __restrict__ a_dst,
                               float* __restrict__ al_s, float* __restrict__ al_d,
                               int n_nodes) {
  int gid = blockIdx.x * blockDim.x + threadIdx.x;
  if (gid >= n_nodes * H_HEADS) return;
  int n = gid >> 2, h = gid & 3;
  const float* row = Hfeat + n * F_DIM + h * C_HID;
  const float* as = a_src + h * C_HID;
  const float* ad = a_dst + h * C_HID;
  float ss = 0.f, sd = 0.f;
#pragma unroll 8
  for (int c = 0; c < C_HID; ++c) {
    float v = row[c];
    ss += v * as[c];
    sd += v * ad[c];
  }
  al_s[gid] = ss;
  al_d[gid] = sd;
}

// sign-aware float atomic max (init must be -inf)
__device__ inline void atomicMaxF(float* addr, float val) {
  if (!(val < 0.0f)) atomicMax((int*)addr, __float_as_int(val));
  else               atomicMin((unsigned int*)addr, __float_as_uint(val));
}

__device__ inline void edge_endpoints(const int* __restrict__ eidx, int ei,
                                      int n_edges, int& s, int& d) {
  if (ei < n_edges) { s = eidx[ei]; d = eidx[n_edges + ei]; }
  else              { s = d = ei - n_edges; }      // self loops appended
}

// e = leaky_relu(al_s[src]+al_d[dst]); store and atomic-max into emax[dst]
__global__ void edge_logit_max_kernel(const int* __restrict__ eidx, int n_edges, int n_nodes,
                                      const float* __restrict__ al_s,
                                      const float* __restrict__ al_d,
                                      float* __restrict__ e_buf,
                                      float* __restrict__ emax) {
  int gid = blockIdx.x * blockDim.x + threadIdx.x;     // edge*4 + head
  int total = (n_edges + n_nodes) * H_HEADS;
  if (gid >= total) return;
  int ei = gid >> 2, h = gid & 3;
  int s, d;
  edge_endpoints(eidx, ei, n_edges, s, d);
  float e = al_s[s * 4 + h] + al_d[d * 4 + h];
  e = (e > 0.f) ? e : NEG_SLOPE * e;
  e_buf[gid] = e;
  atomicMaxF(&emax[d * 4 + h], e);
}

// e_exp = exp(e - emax[dst]); store; atomic-add denom[dst]
__global__ void edge_expsum_kernel(const int* __restrict__ eidx, int n_edges, int n_nodes,
                                   float* __restrict__ e_buf,
                                   const float* __restrict__ emax,
                                   float* __restrict__ denom) {
  int gid = blockIdx.x * blockDim.x + threadIdx.x;
  int total = (n_edges + n_nodes) * H_HEADS;
  if (gid >= total) return;
  int ei = gid >> 2, h = gid & 3;
  int s, d;
  edge_endpoints(eidx, ei, n_edges, s, d);
  float ex = __expf(e_buf[gid] - emax[d * 4 + h]);
  e_buf[gid] = ex;
  atomicAdd(&denom[d * 4 + h], ex);
}

// acc[dst,ch] += h[src,ch] * (e_exp / (denom[dst]+eps)); one thread per (edge,channel)
__global__ void edge_scatter_kernel(const int* __restrict__ eidx, int n_edges, int n_nodes,
                                    const float* __restrict__ Hfeat,
                                    const float* __restrict__ e_buf,
                                    const float* __restrict__ denom,
                                    float* __restrict__ acc) {
  int gid = blockIdx.x * blockDim.x + threadIdx.x;     // edge*128 + ch
  int total = (n_edges + n_nodes) * F_DIM;
  if (gid >= total) return;
  int ei = gid >> 7, ch = gid & 127, h = ch >> 5;
  int s, d;
  edge_endpoints(eidx, ei, n_edges, s, d);
  float alpha = e_buf[ei * 4 + h] / (denom[d * 4 + h] + EPS_V);
  atomicAdd(&acc[d * F_DIM + ch], Hfeat[s * F_DIM + ch] * alpha);
}

// out = leaky_relu(acc + bias), in place
__global__ void finalize_kernel(float* __restrict__ acc, const float* __restrict__ bias,
                                int n_nodes) {
  int gid = blockIdx.x * blockDim.x + threadIdx.x;
  if (gid >= n_nodes * F_DIM) return;
  float v = acc[gid] + bias[gid & 127];
  acc[gid] = (v > 0.f) ? v : NEG_SLOPE * v;
}

// logits = feat @ Wc + bc; log_softmax. One 128-thread block per node.
__global__ void classifier_kernel(const float* __restrict__ feat,
                                  const float* __restrict__ Wc,
                                  const float* __restrict__ bc,
                                  float* __restrict__ out, int n_nodes) {
  __shared__ float row[F_DIM];
  __shared__ float logits[NCLS];
  int n = blockIdx.x;
  int t = threadIdx.x;
  row[t] = feat[n * F_DIM + t];
  __syncthreads();
  if (t < NCLS) {
    float s = bc[t];
#pragma unroll 8
    for (int k = 0; k < F_DIM; ++k) s += row[k] * Wc[k * NCLS + t];
    logits[t] = s;
  }
  __syncthreads();
  if (t < NCLS) {
    float m = -INFINITY;
    for (int j = 0; j < NCLS; ++j) m = fmaxf(m, logits[j]);
    float sum = 0.f;
    for (int j = 0; j < NCLS; ++j) sum += __expf(logits[j] - m);
    out[n * NCLS + t] = logits[t] - m - __logf(sum);
  }
}

// ---------------------------------------------------------------- host
static inline int cdiv(int a, int b) { return (a + b - 1) / b; }

extern "C" void kernel_launch(void* const* d_in, const int* in_sizes, int n_in,
                              void* d_out, int out_size, void* d_ws, size_t ws_size,
                              hipStream_t stream) {
  const float* x    = (const float*)d_in[0];
  const int*   eidx = (const int*)  d_in[1];
  const float* W1   = (const float*)d_in[2];
  const float* a1s  = (const float*)d_in[3];
  const float* a1d  = (const float*)d_in[4];
  const float* b1   = (const float*)d_in[5];
  const float* W2   = (const float*)d_in[6];
  const float* a2s  = (const float*)d_in[7];
  const float* a2d  = (const float*)d_in[8];
  const float* b2   = (const float*)d_in[9];
  const float* Wc   = (const float*)d_in[10];
  const float* bc   = (const float*)d_in[11];
  float* out = (float*)d_out;

  const int N = in_sizes[0] / F_DIM;      // 50000
  const int E = in_sizes[1] / 2;          // 800000
  const int Etot = E + N;

  // workspace carve-up (floats)
  float* bufA  = (float*)d_ws;              // h features   [N*128]
  float* bufB  = bufA  + (size_t)N * F_DIM; // acc / layer output [N*128]
  float* als   = bufB  + (size_t)N * F_DIM; // [N*4]
  float* ald   = als   + (size_t)N * 4;
  float* emax  = ald   + (size_t)N * 4;
  float* denom = emax  + (size_t)N * 4;
  float* ebuf  = denom + (size_t)N * 4;     // [Etot*4]

  const int THR = 256;
  const int nNF = N * F_DIM;
  const int nNH = N * H_HEADS;
  const int nEH = Etot * H_HEADS;
  const int nEF = Etot * F_DIM;

  auto run_layer = [&](const float* inFeat, const float* W, const float* asrc,
                       const float* adst, const float* bias,
                       float* hBuf, float* accBuf) {
    gemm128_wmma<<<cdiv(N, 16), 256, 0, stream>>>(inFeat, W, hBuf, N);
    fill_kernel<<<cdiv(nNF, THR), THR, 0, stream>>>(accBuf, 0.0f, nNF);
    fill_kernel<<<cdiv(nNH, THR), THR, 0, stream>>>(emax, -INFINITY, nNH);
    fill_kernel<<<cdiv(nNH, THR), THR, 0, stream>>>(denom, 0.0f, nNH);
    node_al_kernel<<<cdiv(nNH, THR), THR, 0, stream>>>(hBuf, asrc, adst, als, ald, N);
    edge_logit_max_kernel<<<cdiv(nEH, THR), THR, 0, stream>>>(eidx, E, N, als, ald, ebuf, emax);
    edge_expsum_kernel<<<cdiv(nEH, THR), THR, 0, stream>>>(eidx, E, N, ebuf, emax, denom);
    edge_scatter_kernel<<<cdiv(nEF, THR), THR, 0, stream>>>(eidx, E, N, hBuf, ebuf, denom, accBuf);
    finalize_kernel<<<cdiv(nNF, THR), THR, 0, stream>>>(accBuf, bias, N);
  };

  // layer 1: x -> bufB ; layer 2: bufB -> bufB (GEMM consumes bufB before zero-fill)
  run_layer(x,    W1, a1s, a1d, b1, bufA, bufB);
  run_layer(bufB, W2, a2s, a2d, b2, bufA, bufB);

  classifier_kernel<<<N, F_DIM, 0, stream>>>(bufB, Wc, bc, out, N);
}